// CRFBaseModel_36661840839808
// MI455X (gfx1250) — compile-verified
//
#include <hip/hip_runtime.h>

#define S_LEN 1024
#define BATCH 128
#define NSTATE 101
#define FEAT 64
#define NEGV (-1e4f)
#define NEGBIG (-1e30f)

typedef __attribute__((ext_vector_type(2))) float v2f;
typedef __attribute__((ext_vector_type(8))) float v8f;

__device__ __forceinline__ float lse2(float x, float y) {
    float mx = fmaxf(x, y);
    float mn = fminf(x, y);
    return mx + log1pf(__expf(mn - mx));
}
__device__ __forceinline__ float wave_max(float v) {
#pragma unroll
    for (int off = 16; off > 0; off >>= 1) v = fmaxf(v, __shfl_xor(v, off, 32));
    return v;
}
__device__ __forceinline__ float wave_sum(float v) {
#pragma unroll
    for (int off = 16; off > 0; off >>= 1) v += __shfl_xor(v, off, 32);
    return v;
}
__device__ __forceinline__ int wave_sum_i(int v) {
#pragma unroll
    for (int off = 16; off > 0; off >>= 1) v += __shfl_xor(v, off, 32);
    return v;
}

// ---------------------------------------------------------------------------
// Kernel 1: emissions via V_WMMA_F32_16X16X4_F32.
// em3 = emb[131072x64] @ W[64x3]; keep cols 0,1 (+bias) as packed float2/row.
// W is preloaded into 16 v2f registers with branchless clamped loads, so the
// inner loop is a pure global_load_b64 + v_wmma stream (no EXEC branching).
// Each wave processes 8 consecutive 16-row tiles to amortize the W preload.
// A layout (16x4 f32): lanes 0-15 M=0..15 hold K={0,1}; lanes 16-31 K={2,3}.
// B layout (4x16 f32): lanes 0-15 N, K={0,1}; lanes 16-31 N, K={2,3}.
// D layout: VGPR r = row r (lanes 0-15, N=lane) / row r+8 (lanes 16-31).
// ---------------------------------------------------------------------------
__global__ __launch_bounds__(256) void emis_wmma(const float* __restrict__ emb,
                                                 const float* __restrict__ W,
                                                 const float* __restrict__ bias,
                                                 float* __restrict__ em2) {
    const int lane = threadIdx.x & 31;
    const int wave = threadIdx.x >> 5;
    const int m = lane & 15;   // row of A / col of B
    const int kh = lane >> 4;  // K-half selector
    const int col = (m < 3) ? m : 0;  // clamped: loads always in-bounds
    const bool live = (m < 3);

    v2f wreg[16];
#pragma unroll
    for (int i = 0; i < 16; ++i) {
        const int k = 4 * i + 2 * kh;
        const float wx = W[k * 3 + col];
        const float wy = W[(k + 1) * 3 + col];
        wreg[i].x = live ? wx : 0.0f;
        wreg[i].y = live ? wy : 0.0f;
    }
    const float b0 = bias[0];
    const float b1 = bias[1];

    const int tile0 = (blockIdx.x * 8 + wave) * 8;
    for (int tt = 0; tt < 8; ++tt) {
        const int row0 = (tile0 + tt) * 16;
        const float* arow = emb + (size_t)(row0 + m) * FEAT + 2 * kh;
        v8f c = {};
#pragma unroll
        for (int i = 0; i < 16; ++i) {
            const v2f a = *(const v2f*)(arow + 4 * i);
            c = __builtin_amdgcn_wmma_f32_16x16x4_f32(false, a, false, wreg[i],
                                                      (short)0, c, false, false);
        }
        if (m < 2) {  // lanes 0,1 (rows 0-7) and 16,17 (rows 8-15) hold cols 0,1
            const float bb = (m == 0) ? b0 : b1;
            const int rbase = row0 + (kh ? 8 : 0);
#pragma unroll
            for (int r = 0; r < 8; ++r)
                em2[(size_t)(rbase + r) * 2 + m] = c[r] + bb;
        }
    }
}

// ---------------------------------------------------------------------------
// Kernel 2: forward alpha scan. One wave per batch; lane owns states lane+32c.
// Sparse incoming: j=0/1 <- {0,100}; j=2..98 <- {j-1}; j=99 <- {98,99};
// j=100 <- logsumexp over i=4..99 (butterfly reduction).
// ---------------------------------------------------------------------------
__global__ __launch_bounds__(32) void crf_forward(
    const float* __restrict__ em2, const int* __restrict__ mask,
    const float* __restrict__ start, const float* __restrict__ trans,
    const float* __restrict__ endt, float* __restrict__ alphas,
    float* __restrict__ logZ) {
    const int b = blockIdx.x;
    const int lane = threadIdx.x;

    int len = 0;
    for (int t = lane; t < S_LEN; t += 32) len += mask[b * S_LEN + t];
    len = wave_sum_i(len);

    float wA[4], wB[4], w100[4], enr[4];
    bool is01[4], is100[4], valid[4];
#pragma unroll
    for (int c = 0; c < 4; ++c) {
        const int j = lane + 32 * c;
        valid[c] = (j <= 100);
        is01[c] = (j <= 1);
        is100[c] = (j == 100);
        float a = NEGBIG, bb = NEGBIG;
        if (j == 0) { a = trans[0]; bb = trans[100 * NSTATE + 0]; }
        else if (j == 1) { a = trans[1]; bb = trans[100 * NSTATE + 1]; }
        else if (j == 99) { a = trans[98 * NSTATE + 99]; bb = trans[99 * NSTATE + 99]; }
        else if (j >= 2 && j <= 98) { a = trans[(j - 1) * NSTATE + j]; }
        wA[c] = a;
        wB[c] = bb;
        w100[c] = (j >= 4 && j <= 99) ? trans[j * NSTATE + 100] : NEGBIG;
        enr[c] = !valid[c] ? NEGBIG
                           : (j == 0 ? endt[0] : (j == 100 ? endt[100] : NEGV));
    }

    float a[4];
    {
        const float2 e0 = ((const float2*)em2)[(size_t)b * S_LEN];
#pragma unroll
        for (int c = 0; c < 4; ++c) {
            const int j = lane + 32 * c;
            a[c] = valid[c] ? start[j] + (j == 0 ? e0.x : e0.y) : NEGBIG;
            alphas[((size_t)b * S_LEN) * 128 + c * 32 + lane] = a[c];
        }
    }

    for (int t = 1; t < S_LEN; ++t) {
        const float2 e = ((const float2*)em2)[(size_t)b * S_LEN + t];
        const bool mk = (t < len);
        const float a0b = __shfl(a[0], 0, 32);
        const float a100b = __shfl(a[3], 4, 32);
        const int src = (lane == 0) ? 31 : lane - 1;
        float p[4];
        p[0] = __shfl(a[0], src, 32);
        const float wr1 = __shfl(a[0], 31, 32);
        const float wr2 = __shfl(a[1], 31, 32);
        const float wr3 = __shfl(a[2], 31, 32);
        p[1] = (lane == 0) ? wr1 : __shfl(a[1], src, 32);
        p[2] = (lane == 0) ? wr2 : __shfl(a[2], src, 32);
        p[3] = (lane == 0) ? wr3 : __shfl(a[3], src, 32);

        // 96-way logsumexp feeding state 100
        float v[4], lm = NEGBIG;
#pragma unroll
        for (int c = 0; c < 4; ++c) { v[c] = a[c] + w100[c]; lm = fmaxf(lm, v[c]); }
        const float gm = wave_max(lm);
        float ls = 0.f;
#pragma unroll
        for (int c = 0; c < 4; ++c) ls += __expf(v[c] - gm);
        const float big = gm + __logf(wave_sum(ls));

#pragma unroll
        for (int c = 0; c < 4; ++c) {
            const int j = lane + 32 * c;
            const float in1 = is01[c] ? a0b : p[c];
            const float in2 = is01[c] ? a100b : a[c];
            float nv = lse2(in1 + wA[c], in2 + wB[c]) + (j == 0 ? e.x : e.y);
            if (is100[c]) nv = big + e.y;
            if (!valid[c]) nv = NEGBIG;
            a[c] = mk ? nv : a[c];
            alphas[((size_t)b * S_LEN + t) * 128 + c * 32 + lane] = a[c];
        }
    }

    // logZ = lse_j(alpha_last + end_masked)
    float f[4], lm = NEGBIG;
#pragma unroll
    for (int c = 0; c < 4; ++c) { f[c] = a[c] + enr[c]; lm = fmaxf(lm, f[c]); }
    const float gm = wave_max(lm);
    float ls = 0.f;
#pragma unroll
    for (int c = 0; c < 4; ++c) ls += __expf(f[c] - gm);
    const float gs = wave_sum(ls);
    if (lane == 0) logZ[b] = gm + __logf(gs);
}

// ---------------------------------------------------------------------------
// Kernel 3: backward beta scan fused with marginal softmax output.
// Sparse outgoing: i=0/100 -> {0,1}; i=1..3 -> {i+1}; i=4..98 -> {i+1,100};
// i=99 -> {99,100}. Frozen positions get beta = end-mask vector.
// The 64 MB alpha stream is software-prefetched (global_prefetch).
// ---------------------------------------------------------------------------
__global__ __launch_bounds__(32) void crf_backward(
    const float* __restrict__ em2, const int* __restrict__ mask,
    const float* __restrict__ trans, const float* __restrict__ endt,
    const float* __restrict__ alphas, float* __restrict__ probs_out) {
    const int b = blockIdx.x;
    const int lane = threadIdx.x;

    int len = 0;
    for (int t = lane; t < S_LEN; t += 32) len += mask[b * S_LEN + t];
    len = wave_sum_i(len);

    float wA[4], wB[4], enr[4];
    bool i0100[4], i99[4], valid[4];
#pragma unroll
    for (int c = 0; c < 4; ++c) {
        const int i = lane + 32 * c;
        valid[c] = (i <= 100);
        i0100[c] = (i == 0 || i == 100);
        i99[c] = (i == 99);
        float a = NEGBIG, bb = NEGBIG;
        if (i == 0) { a = trans[0]; bb = trans[1]; }
        else if (i == 100) { a = trans[100 * NSTATE + 0]; bb = trans[100 * NSTATE + 1]; }
        else if (i == 99) { a = trans[99 * NSTATE + 99]; bb = trans[99 * NSTATE + 100]; }
        else if (i >= 1 && i <= 98) {
            a = trans[i * NSTATE + i + 1];
            if (i >= 4) bb = trans[i * NSTATE + 100];
        }
        wA[c] = a;
        wB[c] = bb;
        enr[c] = !valid[c] ? NEGBIG
                           : (i == 0 ? endt[0] : (i == 100 ? endt[100] : NEGV));
    }

    float bt[4];
#pragma unroll
    for (int c = 0; c < 4; ++c) bt[c] = enr[c];

    for (int t = S_LEN - 1; t >= 0; --t) {
        // software prefetch of the alpha row a few steps ahead of the stream
        if (t >= 8)
            __builtin_prefetch(&alphas[((size_t)b * S_LEN + (t - 8)) * 128 + lane * 4],
                               0, 0);
        // probs[t] = softmax_j(alpha[t] + beta[t])
        float s[4], lm = NEGBIG;
#pragma unroll
        for (int c = 0; c < 4; ++c) {
            const float al = alphas[((size_t)b * S_LEN + t) * 128 + c * 32 + lane];
            s[c] = valid[c] ? al + bt[c] : NEGBIG;
            lm = fmaxf(lm, s[c]);
        }
        const float gm = wave_max(lm);
        float ls = 0.f;
#pragma unroll
        for (int c = 0; c < 4; ++c) ls += __expf(s[c] - gm);
        const float inv = 1.0f / wave_sum(ls);
#pragma unroll
        for (int c = 0; c < 4; ++c)
            if (valid[c])
                probs_out[((size_t)b * S_LEN + t) * NSTATE + lane + 32 * c] =
                    __expf(s[c] - gm) * inv;
        if (t == 0) break;

        // beta[t-1] from (e_t, beta[t])
        const float2 e = ((const float2*)em2)[(size_t)b * S_LEN + t];
        const bool mk = (t < len);
        float x[4];
#pragma unroll
        for (int c = 0; c < 4; ++c) {
            const int i = lane + 32 * c;
            x[c] = bt[c] + (i == 0 ? e.x : e.y);
        }
        const float x0b = __shfl(x[0], 0, 32);
        const float x1b = __shfl(x[0], 1, 32);
        const float x100b = __shfl(x[3], 4, 32);
        const int srcn = (lane == 31) ? 0 : lane + 1;
        float q[4];
        q[0] = __shfl(x[0], srcn, 32);
        q[1] = __shfl(x[1], srcn, 32);
        q[2] = __shfl(x[2], srcn, 32);
        q[3] = __shfl(x[3], srcn, 32);
        const float w1 = __shfl(x[1], 0, 32);
        const float w2 = __shfl(x[2], 0, 32);
        const float w3 = __shfl(x[3], 0, 32);
        float xn[4];
        xn[0] = (lane == 31) ? w1 : q[0];
        xn[1] = (lane == 31) ? w2 : q[1];
        xn[2] = (lane == 31) ? w3 : q[2];
        xn[3] = q[3];
#pragma unroll
        for (int c = 0; c < 4; ++c) {
            const float inA = i0100[c] ? x0b : (i99[c] ? x[c] : xn[c]);
            const float inB = i0100[c] ? x1b : x100b;
            const float cand = lse2(inA + wA[c], inB + wB[c]);
            const float nb = mk ? cand : enr[c];
            bt[c] = valid[c] ? nb : NEGBIG;
        }
    }
}

// ---------------------------------------------------------------------------
// Kernel 4: Viterbi forward + LDS backpointer history + serial backtrack.
// 101 KB history lives in LDS (CDNA5 allows 320 KB/workgroup).
// ---------------------------------------------------------------------------
__global__ __launch_bounds__(32) void crf_viterbi(
    const float* __restrict__ em2, const int* __restrict__ mask,
    const float* __restrict__ start, const float* __restrict__ trans,
    const float* __restrict__ endt, const float* __restrict__ logZ,
    float* __restrict__ path_out, float* __restrict__ pp_out) {
    __shared__ unsigned char hist[S_LEN * NSTATE];
    __shared__ unsigned char ptag[S_LEN];
    const int b = blockIdx.x;
    const int lane = threadIdx.x;

    int len = 0;
    for (int t = lane; t < S_LEN; t += 32) len += mask[b * S_LEN + t];
    len = wave_sum_i(len);

    float wA[4], wB[4], w100[4], enr[4];
    int iA[4], iB[4];
    bool is01[4], is100[4], valid[4];
#pragma unroll
    for (int c = 0; c < 4; ++c) {
        const int j = lane + 32 * c;
        valid[c] = (j <= 100);
        is01[c] = (j <= 1);
        is100[c] = (j == 100);
        float a = NEGBIG, bb = NEGBIG;
        int ia = 0, ib = 0;
        if (j == 0) { a = trans[0]; bb = trans[100 * NSTATE + 0]; ia = 0; ib = 100; }
        else if (j == 1) { a = trans[1]; bb = trans[100 * NSTATE + 1]; ia = 0; ib = 100; }
        else if (j == 99) { a = trans[98 * NSTATE + 99]; bb = trans[99 * NSTATE + 99]; ia = 98; ib = 99; }
        else if (j >= 2 && j <= 98) { a = trans[(j - 1) * NSTATE + j]; ia = j - 1; }
        wA[c] = a; wB[c] = bb; iA[c] = ia; iB[c] = ib;
        w100[c] = (j >= 4 && j <= 99) ? trans[j * NSTATE + 100] : NEGBIG;
        enr[c] = !valid[c] ? NEGBIG
                           : (j == 0 ? endt[0] : (j == 100 ? endt[100] : NEGV));
    }

    float s[4];
    {
        const float2 e0 = ((const float2*)em2)[(size_t)b * S_LEN];
#pragma unroll
        for (int c = 0; c < 4; ++c) {
            const int j = lane + 32 * c;
            s[c] = valid[c] ? start[j] + (j == 0 ? e0.x : e0.y) : NEGBIG;
        }
    }

    for (int t = 1; t < S_LEN; ++t) {
        const float2 e = ((const float2*)em2)[(size_t)b * S_LEN + t];
        const bool mk = (t < len);
        const float s0b = __shfl(s[0], 0, 32);
        const float s100b = __shfl(s[3], 4, 32);
        const int src = (lane == 0) ? 31 : lane - 1;
        float p[4];
        p[0] = __shfl(s[0], src, 32);
        const float wr1 = __shfl(s[0], 31, 32);
        const float wr2 = __shfl(s[1], 31, 32);
        const float wr3 = __shfl(s[2], 31, 32);
        p[1] = (lane == 0) ? wr1 : __shfl(s[1], src, 32);
        p[2] = (lane == 0) ? wr2 : __shfl(s[2], src, 32);
        p[3] = (lane == 0) ? wr3 : __shfl(s[3], src, 32);

        // 96-way argmax feeding state 100 (tie -> lower state index)
        float bv = s[0] + w100[0];
        int bi = lane;
#pragma unroll
        for (int c = 1; c < 4; ++c) {
            const float vv = s[c] + w100[c];
            if (vv > bv) { bv = vv; bi = lane + 32 * c; }
        }
#pragma unroll
        for (int off = 16; off > 0; off >>= 1) {
            const float ov = __shfl_xor(bv, off, 32);
            const int oi = __shfl_xor(bi, off, 32);
            if (ov > bv || (ov == bv && oi < bi)) { bv = ov; bi = oi; }
        }

#pragma unroll
        for (int c = 0; c < 4; ++c) {
            const int j = lane + 32 * c;
            const float in1 = is01[c] ? s0b : p[c];
            const float in2 = is01[c] ? s100b : s[c];
            const float cA = in1 + wA[c];
            const float cB = in2 + wB[c];
            float best = cA;
            int bp = iA[c];
            if (cB > best) { best = cB; bp = iB[c]; }
            float nv = best + (j == 0 ? e.x : e.y);
            if (is100[c]) { nv = bv + e.y; bp = bi; }
            int stor;
            if (mk) {
                s[c] = valid[c] ? nv : NEGBIG;
                stor = bp;
            } else {
                stor = j;
            }
            if (valid[c]) hist[t * NSTATE + j] = (unsigned char)stor;
        }
    }

    // final = s + end_masked; global argmax (tie -> lower state)
    float bfv = s[0] + enr[0];
    int bj = lane;
#pragma unroll
    for (int c = 1; c < 4; ++c) {
        const float f = s[c] + enr[c];
        if (f > bfv) { bfv = f; bj = lane + 32 * c; }
    }
#pragma unroll
    for (int off = 16; off > 0; off >>= 1) {
        const float ov = __shfl_xor(bfv, off, 32);
        const int oj = __shfl_xor(bj, off, 32);
        if (ov > bfv || (ov == bfv && oj < bj)) { bfv = ov; bj = oj; }
    }

    if (lane == 0) {
        int tag = bj;
        ptag[S_LEN - 1] = (unsigned char)tag;
        for (int t = S_LEN - 1; t >= 1; --t) {
            tag = hist[t * NSTATE + tag];
            ptag[t - 1] = (unsigned char)tag;
        }
    }
    __syncthreads();
    for (int t = lane; t < S_LEN; t += 32)
        path_out[(size_t)b * S_LEN + t] = (float)ptag[t];
    if (lane == 0) pp_out[b] = __expf(bfv - logZ[b]);
}

extern "C" void kernel_launch(void* const* d_in, const int* in_sizes, int n_in,
                              void* d_out, int out_size, void* d_ws, size_t ws_size,
                              hipStream_t stream) {
    const float* emb = (const float*)d_in[0];    // [B,S,64]
    const int* mask = (const int*)d_in[1];       // [B,S]
    const float* W = (const float*)d_in[2];      // [64,3]
    const float* bias = (const float*)d_in[3];   // [3]
    const float* start = (const float*)d_in[4];  // [101]
    const float* trans = (const float*)d_in[5];  // [101,101]
    const float* endt = (const float*)d_in[6];   // [101]

    float* out = (float*)d_out;
    float* probs_out = out;                                        // B*S*101
    float* path_out = out + (size_t)BATCH * S_LEN * NSTATE;        // B*S
    float* pp_out = path_out + (size_t)BATCH * S_LEN;              // B

    float* ws_em = (float*)d_ws;                                   // B*S*2
    float* ws_alpha = ws_em + (size_t)BATCH * S_LEN * 2;           // B*S*128
    float* ws_logZ = ws_alpha + (size_t)BATCH * S_LEN * 128;       // B

    // 8192 tiles of 16 rows; 8 tiles per wave; 8 waves per block -> 128 blocks
    emis_wmma<<<(BATCH * S_LEN / 16) / 64, 256, 0, stream>>>(emb, W, bias, ws_em);
    crf_forward<<<BATCH, 32, 0, stream>>>(ws_em, mask, start, trans, endt,
                                          ws_alpha, ws_logZ);
    crf_backward<<<BATCH, 32, 0, stream>>>(ws_em, mask, trans, endt, ws_alpha,
                                           probs_out);
    crf_viterbi<<<BATCH, 32, 0, stream>>>(ws_em, mask, start, trans, endt,
                                          ws_logZ, path_out, pp_out);
}